// LearnableSSM_63952063037562
// MI455X (gfx1250) — compile-verified
//
#include <hip/hip_runtime.h>
#include <hip/hip_bf16.h>
#include <stdint.h>

// ---------------- problem constants (match reference) ----------------
#define T_STEPS 2048
#define NPART   16384
#define THREADS 1024
#define PPT     (NPART / THREADS)   // 16 particles per thread
#define CHUNKS  (PPT / 4)           // 4 float4 chunks per thread
#define PROC_STD    0.5f
#define INV_OBS_STD (1.0f / 0.3f)
#define PF_EPS      1e-8f
#define INV_TEMP    10.0f

// ---------------- CDNA5 async global->LDS copy ----------------
typedef int v4i __attribute__((ext_vector_type(4)));
typedef __attribute__((address_space(1))) v4i g_v4i; // global int4
typedef __attribute__((address_space(3))) v4i l_v4i; // LDS int4

#if __has_builtin(__builtin_amdgcn_global_load_async_to_lds_b128)
#define ASYNC_COPY_B128(gsrc, ldst)                                          \
  __builtin_amdgcn_global_load_async_to_lds_b128((g_v4i*)(gsrc),             \
                                                 (l_v4i*)(ldst), 0, 0)
#else
// fallback: emit the CDNA5 instruction directly (vaddr-64 'off' form)
#define ASYNC_COPY_B128(gsrc, ldst)                                          \
  asm volatile("global_load_async_to_lds_b128 %0, %1, off"                   \
               :: "v"((uint32_t)(uintptr_t)(ldst)),                          \
                  "v"((uint64_t)(uintptr_t)(gsrc))                           \
               : "memory")
#endif

#if __has_builtin(__builtin_amdgcn_s_wait_asynccnt)
#define WAIT_ASYNC0() __builtin_amdgcn_s_wait_asynccnt(0)
#else
#define WAIT_ASYNC0() asm volatile("s_wait_asynccnt 0" ::: "memory")
#endif

// ---------------- wave32 reductions ----------------
__device__ __forceinline__ float wred_sum(float v) {
  v += __shfl_xor(v, 16);
  v += __shfl_xor(v, 8);
  v += __shfl_xor(v, 4);
  v += __shfl_xor(v, 2);
  v += __shfl_xor(v, 1);
  return v;
}
__device__ __forceinline__ float wred_max(float v) {
  v = fmaxf(v, __shfl_xor(v, 16));
  v = fmaxf(v, __shfl_xor(v, 8));
  v = fmaxf(v, __shfl_xor(v, 4));
  v = fmaxf(v, __shfl_xor(v, 2));
  v = fmaxf(v, __shfl_xor(v, 1));
  return v;
}

// ---------------- parallel pass: precompute 10*gumbel(u) ----------------
__global__ void pf_gumbel_pre_kernel(const float* __restrict__ u,
                                     float* __restrict__ g10, long n) {
  long i = (long)blockIdx.x * blockDim.x + threadIdx.x;
  const long stride = (long)gridDim.x * blockDim.x;
  for (; i < n; i += stride) {
    g10[i] = -INV_TEMP * __logf(-__logf(u[i] + PF_EPS) + PF_EPS);
  }
}

// ---------------- sequential persistent kernel (one WGP) ----------------
__launch_bounds__(THREADS, 1)
__global__ void pf_seq_kernel(const float* __restrict__ obs,
                              const float* __restrict__ coef_p,
                              const float* __restrict__ init,
                              const float* __restrict__ eps,
                              const float* __restrict__ gsrc, // u or 10*gumbel
                              float* __restrict__ out,
                              int gumbel_pre) {
  // dynamic LDS: double-buffered eps row, 2 * NPART floats = 128 KB
  extern __shared__ float sbuf[];
  __shared__ float obs_s[T_STEPS];           // 8 KB
  __shared__ float redS[32], redM[32], redZ[32], redXE[32];

  const int tid  = threadIdx.x;
  const int lane = tid & 31;
  const int wid  = tid >> 5;
  const float c  = coef_p[0];

  // preload all observations into LDS
  for (int i = tid; i < T_STEPS; i += THREADS) obs_s[i] = obs[i];

  // persistent per-thread particle state: particle index = cc*4096 + tid*4 + j
  float x[PPT], w[PPT];
#pragma unroll
  for (int cc = 0; cc < CHUNKS; ++cc) {
    const float4 v = ((const float4*)init)[cc * THREADS + tid];
    x[4 * cc + 0] = v.x; x[4 * cc + 1] = v.y;
    x[4 * cc + 2] = v.z; x[4 * cc + 3] = v.w;
  }
#pragma unroll
  for (int p = 0; p < PPT; ++p) w[p] = 1.0f / (float)NPART;

  // async-prefetch of an eps row into LDS buffer b (16B per lane, coalesced)
  auto issue_eps = [&](int t, int b) {
    const float* ge = eps + (size_t)t * NPART;
    float* le = sbuf + (size_t)b * NPART;
#pragma unroll
    for (int cc = 0; cc < CHUNKS; ++cc) {
      const int off = cc * THREADS * 4 + tid * 4; // float index
      ASYNC_COPY_B128(ge + off, le + off);
    }
  };

  issue_eps(0, 0);

  for (int t = 0; t < T_STEPS; ++t) {
    WAIT_ASYNC0();        // this wave's async loads for step t complete
    __syncthreads();      // -> every wave's loads complete, buffer readable
    if (t + 1 < T_STEPS) issue_eps(t + 1, (t + 1) & 1);

    // software-pipelined register load of gumbel source row (consumed phase 2)
    const float4* gu4 = (const float4*)(gsrc + (size_t)t * NPART);
    float4 uu[CHUNKS];
#pragma unroll
    for (int cc = 0; cc < CHUNKS; ++cc) uu[cc] = gu4[cc * THREADS + tid];

    const float y = obs_s[t];
    const float4* le4 = (const float4*)(sbuf + (size_t)(t & 1) * NPART);

    // ---- phase 1: transition + likelihood + local sum ----
    float sloc = 0.0f;
#pragma unroll
    for (int cc = 0; cc < CHUNKS; ++cc) {
      const float4 e4 = le4[cc * THREADS + tid];
      const float ev[4] = {e4.x, e4.y, e4.z, e4.w};
#pragma unroll
      for (int j = 0; j < 4; ++j) {
        const int p = 4 * cc + j;
        const float xp = __builtin_fmaf(c, x[p], PROC_STD * ev[j]);
        x[p] = xp;
        const float d = (y - xp) * INV_OBS_STD;
        const float wp = w[p] * __expf(-0.5f * d * d);
        w[p] = wp;
        sloc += wp;
      }
    }
    // reduction 1: S = sum(w*lik)
    // stage A: wave partials to LDS; stage B: lane-indexed read + shuffle tree
    {
      const float vs = wred_sum(sloc);
      if (lane == 0) redS[wid] = vs;
    }
    __syncthreads();
    const float S = wred_sum(redS[lane]);   // identical in all lanes/waves
    // softmax shift-invariance: log(w/D+eps) = log(w + eps*D) - log D,
    // D = S + N*eps; the -log D constant cancels in the softmax.
    const float cadd = PF_EPS * (S + (float)NPART * PF_EPS);

    // ---- phase 2: logits (w[] becomes logit) + local max ----
    float mloc = -3.4e38f;
#pragma unroll
    for (int cc = 0; cc < CHUNKS; ++cc) {
      const float uv[4] = {uu[cc].x, uu[cc].y, uu[cc].z, uu[cc].w};
#pragma unroll
      for (int j = 0; j < 4; ++j) {
        const int p = 4 * cc + j;
        const float g10 = gumbel_pre
            ? uv[j]
            : -INV_TEMP * __logf(-__logf(uv[j] + PF_EPS) + PF_EPS);
        const float l = __builtin_fmaf(INV_TEMP, __logf(w[p] + cadd), g10);
        w[p] = l;
        mloc = fmaxf(mloc, l);
      }
    }
    // reduction 2: M = max(logit)
    {
      const float vm = wred_max(mloc);
      if (lane == 0) redM[wid] = vm;
    }
    __syncthreads();
    const float M = wred_max(redM[lane]);

    // ---- phase 3: softmax + estimate (fused Z and X·E sums) ----
    float zloc = 0.0f, xeloc = 0.0f;
#pragma unroll
    for (int p = 0; p < PPT; ++p) {
      const float e = __expf(w[p] - M);
      w[p] = e;
      zloc += e;
      xeloc = __builtin_fmaf(x[p], e, xeloc);
    }
    // reduction 3: Z = sum(e), XE = sum(x*e)
    {
      const float vz = wred_sum(zloc);
      const float vxe = wred_sum(xeloc);
      if (lane == 0) { redZ[wid] = vz; redXE[wid] = vxe; }
    }
    __syncthreads();
    const float Z  = wred_sum(redZ[lane]);
    const float XE = wred_sum(redXE[lane]);
    const float invZ = 1.0f / Z;
#pragma unroll
    for (int p = 0; p < PPT; ++p) w[p] *= invZ; // next-step weights
    if (tid == 0) out[t] = XE * invZ;
  }
}

// ---------------- host launcher ----------------
extern "C" void kernel_launch(void* const* d_in, const int* in_sizes, int n_in,
                              void* d_out, int out_size, void* d_ws, size_t ws_size,
                              hipStream_t stream) {
  (void)in_sizes; (void)n_in; (void)out_size;
  const float* obs  = (const float*)d_in[0];
  const float* coef = (const float*)d_in[1];
  const float* init = (const float*)d_in[2];
  const float* eps  = (const float*)d_in[3];
  const float* u    = (const float*)d_in[4];
  float* out = (float*)d_out;

  const long total = (long)T_STEPS * (long)NPART;
  const int use_pre = (ws_size >= (size_t)total * sizeof(float)) ? 1 : 0;
  const float* gsrc = u;
  if (use_pre) {
    float* g10 = (float*)d_ws;
    pf_gumbel_pre_kernel<<<8192, 256, 0, stream>>>(u, g10, total);
    gsrc = g10;
  }
  const size_t smem = (size_t)2 * NPART * sizeof(float); // 128 KB ping-pong
  pf_seq_kernel<<<1, THREADS, smem, stream>>>(obs, coef, init, eps, gsrc, out,
                                              use_pre);
}